// KeyDecider_37907381354767
// MI455X (gfx1250) — compile-verified
//
#include <hip/hip_runtime.h>
#include <math.h>

// Problem constants (x: (64, 34, 256, 256) fp32; k = 17; N = 65536; W=H=IMG=256)
#define BATCH   64
#define CCH     34
#define KCH     17
#define NPOS    65536
#define TPB     256
#define CHUNK   8
#define ITERS   32   // NPOS / (TPB * CHUNK)

typedef float f4  __attribute__((ext_vector_type(4)));
typedef float v2f __attribute__((ext_vector_type(2)));
typedef float v8f __attribute__((ext_vector_type(8)));

// Exact fp32 sum of one value per lane across a wave32, using
// V_WMMA_F32_16X16X4_F32: A = (x, 0) per lane -> A[m][0]=x_m, A[m][2]=x_{m+16},
// B = ones(4x16) => D[m][n] = x_m + x_{m+16}. Each lane then holds 8 row-sums
// in its D registers; summing them gives rows 0-7 (lanes 0-15) or 8-15
// (lanes 16-31); one shfl_xor(16) finishes the exact 32-lane sum.
__device__ __forceinline__ float wave_sum32_wmma(float x) {
    v2f a; a.x = x;    a.y = 0.0f;
    v2f b; b.x = 1.0f; b.y = 1.0f;
    v8f c = {};
    v8f d = __builtin_amdgcn_wmma_f32_16x16x4_f32(
        /*neg_a=*/false, a, /*neg_b=*/false, b,
        /*c_mod=*/(short)0, c, /*reuse_a=*/false, /*reuse_b=*/false);
    float s = ((d[0] + d[1]) + (d[2] + d[3])) + ((d[4] + d[5]) + (d[6] + d[7]));
    s += __shfl_xor(s, 16, 32);
    return s;
}

__global__ __launch_bounds__(TPB)
void KeyDecider_37907381354767_kernel(const float* __restrict__ x,
                                      float* __restrict__ out) {
    const int row = blockIdx.x;            // 0 .. B*KCH-1
    const int b   = row / KCH;
    const int c   = row % KCH;
    const float* __restrict__ hp = x + (size_t)(b * CCH + c)       * (size_t)NPOS;
    const float* __restrict__ vp = x + (size_t)(b * CCH + KCH + c) * (size_t)NPOS;
    const int tid = threadIdx.x;

    // Online softmax accumulators (per thread, over 256 strided elements)
    float m  = -INFINITY;
    float Z  = 0.0f;   // sum of exp(h - m)
    float Sp = 0.0f;   // sum of exp(h - m) * p
    float Sv = 0.0f;   // sum of exp(h - m) * v

    for (int it = 0; it < ITERS; ++it) {
        const int base = (it * TPB + tid) * CHUNK;

        // Speculative prefetch ~4 iterations (32KB/stream) ahead -> global_prefetch_b8
        const int pf = base + 4 * TPB * CHUNK;
        if (pf < NPOS) {
            __builtin_prefetch(hp + pf, 0, 0);
            __builtin_prefetch(vp + pf, 0, 0);
        }

        // Stream once, bypass cache retention: non-temporal b128 loads
        f4 h0 = __builtin_nontemporal_load((const f4*)(hp + base));
        f4 h1 = __builtin_nontemporal_load((const f4*)(hp + base + 4));
        f4 v0 = __builtin_nontemporal_load((const f4*)(vp + base));
        f4 v1 = __builtin_nontemporal_load((const f4*)(vp + base + 4));

        // Chunk max -> single rescale exp per 8 elements (1.125 exp/elem total)
        float cm = fmaxf(fmaxf(fmaxf(h0.x, h0.y), fmaxf(h0.z, h0.w)),
                         fmaxf(fmaxf(h1.x, h1.y), fmaxf(h1.z, h1.w)));
        float nm = fmaxf(m, cm);
        float sc = __expf(m - nm);          // exp(-inf)=0 on first iteration
        Z *= sc; Sp *= sc; Sv *= sc;

        const float pb = (float)base;       // positions are exact in fp32 (< 2^24)
        float e;
        e = __expf(h0.x - nm); Z += e; Sp = fmaf(e, pb + 0.0f, Sp); Sv = fmaf(e, v0.x, Sv);
        e = __expf(h0.y - nm); Z += e; Sp = fmaf(e, pb + 1.0f, Sp); Sv = fmaf(e, v0.y, Sv);
        e = __expf(h0.z - nm); Z += e; Sp = fmaf(e, pb + 2.0f, Sp); Sv = fmaf(e, v0.z, Sv);
        e = __expf(h0.w - nm); Z += e; Sp = fmaf(e, pb + 3.0f, Sp); Sv = fmaf(e, v0.w, Sv);
        e = __expf(h1.x - nm); Z += e; Sp = fmaf(e, pb + 4.0f, Sp); Sv = fmaf(e, v1.x, Sv);
        e = __expf(h1.y - nm); Z += e; Sp = fmaf(e, pb + 5.0f, Sp); Sv = fmaf(e, v1.y, Sv);
        e = __expf(h1.z - nm); Z += e; Sp = fmaf(e, pb + 6.0f, Sp); Sv = fmaf(e, v1.z, Sv);
        e = __expf(h1.w - nm); Z += e; Sp = fmaf(e, pb + 7.0f, Sp); Sv = fmaf(e, v1.w, Sv);
        m = nm;
    }

    // Block reduction: 256 per-thread tuples -> LDS
    __shared__ float sm_[TPB], sz_[TPB], sp_[TPB], sv_[TPB];
    sm_[tid] = m; sz_[tid] = Z; sp_[tid] = Sp; sv_[tid] = Sv;
    __syncthreads();

    if (tid < 32) {                        // wave 0 only: EXEC all-ones (WMMA-legal)
        // Block max over all 256 partial maxima
        float M = -INFINITY;
        #pragma unroll
        for (int i = 0; i < 8; ++i) M = fmaxf(M, sm_[tid + 32 * i]);
        #pragma unroll
        for (int off = 16; off > 0; off >>= 1) M = fmaxf(M, __shfl_xor(M, off, 32));

        // Rescale each partial to the common max, 8 tuples per lane
        float z = 0.0f, sp = 0.0f, sv = 0.0f;
        #pragma unroll
        for (int i = 0; i < 8; ++i) {
            const int t = tid + 32 * i;     // conflict-free LDS banks
            const float s = __expf(sm_[t] - M);
            z  = fmaf(sz_[t], s, z);
            sp = fmaf(sp_[t], s, sp);
            sv = fmaf(sv_[t], s, sv);
        }

        // Exact fp32 cross-lane sums via V_WMMA_F32_16X16X4_F32
        const float Zt  = wave_sum32_wmma(z);
        const float Spt = wave_sum32_wmma(sp);
        const float Svt = wave_sum32_wmma(sv);

        if (tid == 0) {
            const float kif = rintf(Spt / Zt);            // jnp.round == RNE
            out[row * 3 + 0] = fmodf(kif, 256.0f);        // mod(ki,W)/W*IMG_W, W==IMG_W
            out[row * 3 + 1] = floorf(kif * (1.0f / 256.0f)); // floor(ki/W)/H*IMG_H
            out[row * 3 + 2] = Svt / Zt;
        }
    }
}

extern "C" void kernel_launch(void* const* d_in, const int* in_sizes, int n_in,
                              void* d_out, int out_size, void* d_ws, size_t ws_size,
                              hipStream_t stream) {
    const float* x = (const float*)d_in[0];
    float* out = (float*)d_out;
    dim3 grid(BATCH * KCH);   // 1088 blocks, one (b,c) row each
    dim3 block(TPB);          // 8 wave32s
    KeyDecider_37907381354767_kernel<<<grid, block, 0, stream>>>(x, out);
}